// GNNLayers_71768903516463
// MI455X (gfx1250) — compile-verified
//
#include <hip/hip_runtime.h>

typedef __attribute__((ext_vector_type(2))) float v2f;
typedef __attribute__((ext_vector_type(8))) float v8f;

#define NNODES  (128 * 512)       // 65536 nodes per side
#define NEDGES  (128 * 512 * 16)  // 1,048,576 edges per side
#define NGRAPHS 128
#define NPG     512               // nodes per graph
#define EPG     (512 * 16)        // edges per graph
#define RES     128               // output resolution / pooled rows per graph

// ---------------------------------------------------------------------------
// Per-graph degree -> dinv = rsqrt(1 + indeg), counted in LDS (dst is always
// inside its own graph, so local index = dst & 511).
// ---------------------------------------------------------------------------
__global__ void __launch_bounds__(512) k_dinv(const int* __restrict__ ei,
                                              float* __restrict__ dinv) {
  __shared__ int cnt[NPG];
  const int b = blockIdx.x;
  const int tid = threadIdx.x;
  if (tid < NPG) cnt[tid] = 1;  // self-loop
  __syncthreads();
  for (int e = tid; e < EPG; e += blockDim.x)
    atomicAdd(&cnt[ei[NEDGES + b * EPG + e] & (NPG - 1)], 1);
  __syncthreads();
  if (tid < NPG) dinv[b * NPG + tid] = rsqrtf((float)cnt[tid]);
}

// ---------------------------------------------------------------------------
// C[M, NT*16] = A[M,K] @ B[K, NT*16] row-major, fp32 WMMA 16x16x4.
// The weight matrix is staged in LDS pair-interleaved along K so each
// B-fragment ({B[k][c], B[k+1][c]}) is a single ds_load_b64; A-fragments are
// single global_load_b64. One wave per 16-row M-tile, NT accumulators reuse
// the A fragment. All strides compile-time.
// ---------------------------------------------------------------------------
template <int NT, int K>
__global__ void __launch_bounds__(256) k_gemm_wmma(const float* __restrict__ A,
                                                   const float* __restrict__ B,
                                                   float* __restrict__ C, int M) {
  const int N = NT * 16;
  __shared__ v2f sB[(K / 2) * N];  // sB[kp*N+c] = {B[2kp][c], B[2kp+1][c]}
  for (int i = threadIdx.x; i < (K / 2) * N; i += blockDim.x) {
    int kp = i / N;
    int c  = i - kp * N;
    v2f w;
    w.x = B[(size_t)(2 * kp) * N + c];
    w.y = B[(size_t)(2 * kp + 1) * N + c];
    sB[i] = w;
  }
  __syncthreads();

  int wave = (int)((blockIdx.x * blockDim.x + threadIdx.x) >> 5);
  int lane = threadIdx.x & 31;
  if (wave >= (M >> 4)) return;  // wave-uniform; EXEC all-1s at WMMA
  int r  = lane & 15;
  int kb = (lane >> 4) << 1;  // 0 or 2
  const float* Arow = A + (size_t)(wave * 16 + r) * K;

  v8f acc[NT] = {};
  for (int k = 0; k < K; k += 4) {
    v2f a = *(const v2f*)(Arow + k + kb);       // one global_load_b64
    v2f bf[NT];
#pragma unroll
    for (int t = 0; t < NT; ++t)                // NT x ds_load_b64, const offsets
      bf[t] = sB[((k + kb) >> 1) * N + t * 16 + r];
#pragma unroll
    for (int t = 0; t < NT; ++t)
      acc[t] = __builtin_amdgcn_wmma_f32_16x16x4_f32(false, a, false, bf[t], (short)0,
                                                     acc[t], false, false);
  }
  int rbase = (lane >> 4) << 3;
#pragma unroll
  for (int t = 0; t < NT; ++t) {
    float* Crow = C + (size_t)(wave * 16 + rbase) * N + t * 16 + r;
#pragma unroll
    for (int v = 0; v < 8; ++v) Crow[(size_t)v * N] = acc[t][v];
  }
}

// ---------------------------------------------------------------------------
// Fused GCN aggregation, one workgroup per graph, everything LDS-resident:
//   hl  = h tile (512 x C)           (each element reused ~DEG=16 times)
//   acc = self term, then ds_add_f32 scatter over the graph's 8192 edges
//   -> y = relu(acc + bias)  (next layer input)
//   -> pool[b][r][:] = 0.5*(relu(acc[4r+1]+b) + relu(acc[4r+2]+b))
// LDS: 2*512*C + 512 floats  (C=64: 258 KB < 320 KB per WGP).
// ---------------------------------------------------------------------------
template <int C>
__global__ void __launch_bounds__(512) k_agg(
    const float* __restrict__ h, const int* __restrict__ ei,
    const float* __restrict__ dinv, const float* __restrict__ bias,
    float* __restrict__ y, float* __restrict__ pout) {
  extern __shared__ float smem[];
  float* acc = smem;                 // 512*C
  float* hl  = smem + NPG * C;       // 512*C
  float* dv  = smem + 2 * NPG * C;   // 512
  const int b = blockIdx.x;
  const int gbase = b * NPG;
  const int tid = threadIdx.x;

  for (int i = tid; i < NPG; i += blockDim.x) dv[i] = dinv[gbase + i];
  for (int idx = tid; idx < NPG * C; idx += blockDim.x)
    hl[idx] = h[(size_t)gbase * C + idx];
  __syncthreads();

  // self-loop term: acc[i][c] = h[i][c] * dinv[i]^2
  for (int idx = tid; idx < NPG * C; idx += blockDim.x) {
    float d = dv[idx / C];
    acc[idx] = hl[idx] * d * d;
  }
  __syncthreads();

  // edge scatter, 4 channels per thread-task, all in LDS
  const int chunks = C / 4;
  for (int t = tid; t < EPG * chunks; t += blockDim.x) {
    int e  = t / chunks;
    int cg = (t - e * chunks) * 4;
    int s  = ei[(size_t)b * EPG + e] - gbase;
    int d  = ei[(size_t)NEDGES + (size_t)b * EPG + e] - gbase;
    float w = dv[s] * dv[d];
    float* ad = acc + d * C + cg;
    const float* hs = hl + s * C + cg;
    atomicAdd(&ad[0], hs[0] * w);
    atomicAdd(&ad[1], hs[1] * w);
    atomicAdd(&ad[2], hs[2] * w);
    atomicAdd(&ad[3], hs[3] * w);
  }
  __syncthreads();

  // bias + relu -> next-layer activations
  for (int idx = tid; idx < NPG * C; idx += blockDim.x) {
    float v = acc[idx] + bias[idx & (C - 1)];
    y[(size_t)gbase * C + idx] = fmaxf(v, 0.0f);
  }
  // fused bilinear pooling (reads finalized acc; no extra barrier needed)
  for (int idx = tid; idx < RES * C; idx += blockDim.x) {
    int c = idx & (C - 1);
    int r = idx / C;
    float bb = bias[c];
    float v1 = fmaxf(acc[(4 * r + 1) * C + c] + bb, 0.0f);
    float v2 = fmaxf(acc[(4 * r + 2) * C + c] + bb, 0.0f);
    pout[(size_t)b * RES * C + idx] = 0.5f * (v1 + v2);
  }
}

// ---------------------------------------------------------------------------
// Batched similarity: out[b] = U[b] (128 x D) @ V[b]^T, fp32 WMMA.
// One 256-thread block (8 waves) per graph; the V tile (128 x D <= 32 KB) is
// staged in LDS and shared by all 8 waves. B-fragments {V[n][k],V[n][k+1]}
// are contiguous -> single ds_load_b64 each, constant offsets (D templated).
// Each wave owns one 16-row tile and all 8 column tiles (A-fragment reuse).
// ---------------------------------------------------------------------------
template <int D>
__global__ void __launch_bounds__(256) k_sim_wmma(const float* __restrict__ U,
                                                  const float* __restrict__ V,
                                                  float* __restrict__ out) {
  __shared__ float sV[RES * D];  // <= 32 KB
  const int b = blockIdx.x;
  const float* Vb = V + (size_t)b * RES * D;
  for (int i = threadIdx.x; i < RES * D; i += blockDim.x) sV[i] = Vb[i];
  __syncthreads();

  int tm   = threadIdx.x >> 5;  // wave -> output tile row
  int lane = threadIdx.x & 31;
  int r    = lane & 15;
  int kb   = (lane >> 4) << 1;
  const float* Ar = U + ((size_t)b * RES + tm * 16 + r) * D;

  v8f acc[8] = {};
  for (int k = 0; k < D; k += 4) {
    v2f a = *(const v2f*)(Ar + k + kb);  // one global_load_b64
    v2f bf[8];
#pragma unroll
    for (int t = 0; t < 8; ++t)          // 8 x ds_load_b64, const offsets
      bf[t] = *(const v2f*)(sV + (t * 16 + r) * D + k + kb);
#pragma unroll
    for (int t = 0; t < 8; ++t)
      acc[t] = __builtin_amdgcn_wmma_f32_16x16x4_f32(false, a, false, bf[t], (short)0,
                                                     acc[t], false, false);
  }
  int rbase = (lane >> 4) << 3;
#pragma unroll
  for (int t = 0; t < 8; ++t) {
    float* Co = out + (size_t)b * RES * RES + (size_t)(tm * 16 + rbase) * RES + t * 16 + r;
#pragma unroll
    for (int v = 0; v < 8; ++v) Co[(size_t)v * RES] = acc[t][v];
  }
}

// ---------------------------------------------------------------------------
extern "C" void kernel_launch(void* const* d_in, const int* in_sizes, int n_in,
                              void* d_out, int out_size, void* d_ws, size_t ws_size,
                              hipStream_t stream) {
  (void)in_sizes; (void)n_in; (void)out_size; (void)ws_size;
  const float* x[2]  = {(const float*)d_in[0], (const float*)d_in[1]};
  const int*   ei[2] = {(const int*)d_in[2],  (const int*)d_in[3]};
  const float* W[3]  = {(const float*)d_in[6], (const float*)d_in[8], (const float*)d_in[10]};
  const float* bv[3] = {(const float*)d_in[7], (const float*)d_in[9], (const float*)d_in[11]};
  float* out = (float*)d_out;

  // Workspace (floats): dinv[2] | B0 | B1 | B2 | pooled P[2][3]  (~57 MB)
  float* ws = (float*)d_ws;
  size_t o = 0;
  float* dinvA = ws + o; o += (size_t)2 * NNODES;
  float* B0 = ws + o; o += (size_t)NNODES * 64;
  float* B1 = ws + o; o += (size_t)NNODES * 64;
  float* B2 = ws + o; o += (size_t)NNODES * 32;
  float* P[2][3];
  const int dims[3] = {64, 32, 16};
  for (int s = 0; s < 2; ++s)
    for (int l = 0; l < 3; ++l) { P[s][l] = ws + o; o += (size_t)NGRAPHS * RES * dims[l]; }

  const unsigned gemmBlocks = (NNODES / 16) * 32 / 256;  // 512
  auto shbytes = [](int C) { return (size_t)(2 * NPG * C + NPG) * sizeof(float); };

  for (int s = 0; s < 2; ++s) {
    float* dv = dinvA + (size_t)s * NNODES;
    k_dinv<<<NGRAPHS, 512, 0, stream>>>(ei[s], dv);

    // Layer 1: 128 -> 64
    k_gemm_wmma<4, 128><<<gemmBlocks, 256, 0, stream>>>(x[s], W[0], B0, NNODES);
    k_agg<64><<<NGRAPHS, 512, shbytes(64), stream>>>(B0, ei[s], dv, bv[0], B1, P[s][0]);
    // Layer 2: 64 -> 32
    k_gemm_wmma<2, 64><<<gemmBlocks, 256, 0, stream>>>(B1, W[1], B0, NNODES);
    k_agg<32><<<NGRAPHS, 512, shbytes(32), stream>>>(B0, ei[s], dv, bv[1], B2, P[s][1]);
    // Layer 3: 32 -> 16
    k_gemm_wmma<1, 32><<<gemmBlocks, 256, 0, stream>>>(B2, W[2], B0, NNODES);
    k_agg<16><<<NGRAPHS, 512, shbytes(16), stream>>>(B0, ei[s], dv, bv[2], B1, P[s][2]);
  }

  // Similarity GEMMs: layer l output at l * 128*128*128
  k_sim_wmma<64><<<NGRAPHS, 256, 0, stream>>>(P[0][0], P[1][0], out);
  k_sim_wmma<32><<<NGRAPHS, 256, 0, stream>>>(P[0][1], P[1][1],
                                              out + (size_t)1 * NGRAPHS * RES * RES);
  k_sim_wmma<16><<<NGRAPHS, 256, 0, stream>>>(P[0][2], P[1][2],
                                              out + (size_t)2 * NGRAPHS * RES * RES);
}